// ResNormedMPNN_62337155334683
// MI455X (gfx1250) — compile-verified
//
#include <hip/hip_runtime.h>

// Problem constants (match reference)
#define NN   50000
#define EE   400000
#define EE2  450000      // E + N self loops
#define CC   64
#define HH   4
#define HC   256         // H*C
#define DEE  16
#define BB   64
#define EPSV 1e-5f
#define SLOPE 0.2f

typedef float v2f __attribute__((ext_vector_type(2)));
typedef float v8f __attribute__((ext_vector_type(8)));

// ---------------------------------------------------------------- utilities
__device__ __forceinline__ void atomicMaxFloat(float* addr, float val) {
  // sign-aware trick; works with -inf initialization
  if (val >= 0.0f) atomicMax((int*)addr, __float_as_int(val));
  else             atomicMin((unsigned int*)addr, __float_as_uint(val));
}

__global__ void fill_neg_inf_kernel(float* p, int n) {
  int i = blockIdx.x * 256 + threadIdx.x;
  if (i < n) p[i] = __int_as_float(0xFF800000); // -inf
}

// -------------------------------------------------- edge_attr mean (sums)
__global__ void ea_sum_kernel(const float* __restrict__ ea, float* __restrict__ easum) {
  __shared__ float red[256];
  int t = threadIdx.x;
  int k = t & 15;
  int row0 = blockIdx.x * 16 + (t >> 4);
  float s = 0.0f;
  for (int r = row0; r < EE; r += gridDim.x * 16)
    s += ea[(size_t)r * DEE + k];
  red[t] = s;
  __syncthreads();
  if (t < 16) {
    float tot = 0.0f;
    #pragma unroll
    for (int i = 0; i < 16; ++i) tot += red[t + 16 * i];
    atomicAdd(&easum[t], tot);
  }
}

// -------------------------------------------------- GraphNorm statistics
__global__ void stats_kernel(const float* __restrict__ xin,
                             const long long* __restrict__ batch,
                             float* __restrict__ S1, float* __restrict__ S2,
                             float* __restrict__ cnt) {
  int t = threadIdx.x;
  int c = t & 63;
  int n = blockIdx.x * 4 + (t >> 6);        // N divisible by 4
  float v = xin[(size_t)n * CC + c];
  int b = (int)batch[n];
  atomicAdd(&S1[b * CC + c], v);
  atomicAdd(&S2[b * CC + c], v * v);
  if (c == 0) atomicAdd(&cnt[b], 1.0f);
}

__global__ void finalize_stats_kernel(const float* __restrict__ S1,
                                      const float* __restrict__ S2,
                                      const float* __restrict__ cnt,
                                      const float* __restrict__ gnw,
                                      const float* __restrict__ gnms,
                                      float* __restrict__ Am, float* __restrict__ Bc) {
  int i = blockIdx.x * 256 + threadIdx.x;
  if (i >= BB * CC) return;
  int b = i >> 6, c = i & 63;
  float nb   = fmaxf(cnt[b], 1.0f);
  float mean = S1[i] / nb;
  float ex2  = S2[i] / nb;
  float ms   = gnms[c];
  // var = E[(x - ms*mean)^2] = E[x^2] - 2*ms*mean*E[x] + ms^2*mean^2
  float var = ex2 - 2.0f * ms * mean * mean + ms * ms * mean * mean;
  Am[i] = ms * mean;
  Bc[i] = rsqrtf(var + EPSV) * gnw[c];
}

__global__ void normalize_kernel(const float* __restrict__ xin,
                                 const long long* __restrict__ batch,
                                 const float* __restrict__ Am,
                                 const float* __restrict__ Bc,
                                 const float* __restrict__ gnb,
                                 float* __restrict__ h) {
  int t = threadIdx.x;
  int c = t & 63;
  int n = blockIdx.x * 4 + (t >> 6);
  int b = (int)batch[n];
  float v = (xin[(size_t)n * CC + c] - Am[b * CC + c]) * Bc[b * CC + c] + gnb[c];
  h[(size_t)n * CC + c] = v > 0.0f ? v : 0.0f;   // ReLU
}

// -------------------------------------------------- node GEMMs via fp32 WMMA
// out[N,256] = h[N,64] @ W[64,256], one 16x16 tile per wave, K stepped by 4.
__global__ void __launch_bounds__(128)
gemm_node_kernel(const float* __restrict__ h,
                 const float* __restrict__ Wl, const float* __restrict__ Wr,
                 float* __restrict__ xl, float* __restrict__ xr) {
  const float* W  = blockIdx.z ? Wr : Wl;
  float*       out = blockIdx.z ? xr : xl;
  int lane = threadIdx.x;            // 0..31
  int lm = lane & 15, lh = lane >> 4;
  int m0 = blockIdx.x * 16;                                  // 3125 tiles, exact
  int n0 = (blockIdx.y * 4 + threadIdx.y) * 16;              // 16 n-tiles
  const float* hrow = h + (size_t)(m0 + lm) * CC;
  v8f acc = {};
  #pragma unroll
  for (int k0 = 0; k0 < CC; k0 += 4) {
    int ka = k0 + 2 * lh;                     // A: lanes0-15 -> K{k0,k0+1}, lanes16-31 -> K{k0+2,k0+3}
    v2f a; a.x = hrow[ka];                  a.y = hrow[ka + 1];
    v2f b; b.x = W[(size_t)ka * HC + n0 + lm]; b.y = W[(size_t)(ka + 1) * HC + n0 + lm];
    acc = __builtin_amdgcn_wmma_f32_16x16x4_f32(false, a, false, b, (short)0, acc,
                                                false, false);
  }
  #pragma unroll
  for (int i = 0; i < 8; ++i)                 // C layout: VGPR i -> M = i + 8*lh
    out[(size_t)(m0 + i + 8 * lh) * HC + n0 + lm] = acc[i];
}

// -------------------------------------------------- edge pass A: logits + segment max
__global__ void __launch_bounds__(256)
edge_logits_kernel(const long long* __restrict__ eidx,
                   const float* __restrict__ ea,
                   const float* __restrict__ easum,
                   const float* __restrict__ xl, const float* __restrict__ xr,
                   const float* __restrict__ WeL, const float* __restrict__ attL,
                   float* __restrict__ logitBuf, float* __restrict__ mbuf) {
  __shared__ float sWe[DEE * HC];   // 16 KB
  __shared__ float sAtt[HC];        // 1 KB
  int t = threadIdx.y * 32 + threadIdx.x;
  for (int i = t; i < DEE * HC; i += 256) sWe[i] = WeL[i];
  if (t < HC) sAtt[t] = attL[t];
  __syncthreads();

  int e = blockIdx.x * 8 + threadIdx.y;       // E2 divisible by 8
  int lane = threadIdx.x;
  long long s, d; float eav;
  if (e < EE) {
    s = eidx[e]; d = eidx[EE + e];
    eav = ea[(size_t)e * DEE + (lane & 15)];
  } else {
    s = e - EE; d = s;
    eav = easum[lane & 15] * (1.0f / (float)EE);   // fill_value='mean'
  }
  const float* xls = xl + (size_t)s * HC;
  const float* xrd = xr + (size_t)d * HC;

  float ph[HH] = {0.f, 0.f, 0.f, 0.f};
  #pragma unroll
  for (int j = 0; j < 8; ++j) {
    int ch = lane + 32 * j;                   // head = j>>1 for all lanes
    float ee = 0.0f;
    #pragma unroll
    for (int k = 0; k < DEE; ++k) {
      float eak = __shfl(eav, k, 32);
      ee += eak * sWe[k * HC + ch];
    }
    float f = xls[ch] + xrd[ch] + ee;
    f = f > 0.0f ? f : SLOPE * f;             // leaky relu
    ph[j >> 1] += f * sAtt[ch];
  }
  #pragma unroll
  for (int off = 16; off > 0; off >>= 1) {
    #pragma unroll
    for (int hh = 0; hh < HH; ++hh) ph[hh] += __shfl_xor(ph[hh], off, 32);
  }
  if (lane == 0) {
    #pragma unroll
    for (int hh = 0; hh < HH; ++hh) {
      logitBuf[(size_t)e * HH + hh] = ph[hh];
      atomicMaxFloat(&mbuf[d * HH + hh], ph[hh]);
    }
  }
}

// -------------------------------------------------- edge pass B: exp + denom
__global__ void edge_exp_kernel(const long long* __restrict__ eidx,
                                const float* __restrict__ mbuf,
                                float* __restrict__ aBuf,      // in: logits, out: a
                                float* __restrict__ denom) {
  int idx = blockIdx.x * 256 + threadIdx.x;
  if (idx >= EE2 * HH) return;
  int e = idx >> 2, hh = idx & 3;
  long long d = (e < EE) ? eidx[EE + e] : (long long)(e - EE);
  float a = __expf(aBuf[idx] - mbuf[d * HH + hh]);
  aBuf[idx] = a;
  atomicAdd(&denom[d * HH + hh], a);
}

// -------------------------------------------------- edge pass C: weighted scatter
__global__ void __launch_bounds__(256)
edge_aggregate_kernel(const long long* __restrict__ eidx,
                      const float* __restrict__ aBuf,
                      const float* __restrict__ denom,
                      const float* __restrict__ xl,
                      float* __restrict__ accOut) {
  int e = blockIdx.x * 8 + threadIdx.y;
  int lane = threadIdx.x;
  long long s, d;
  if (e < EE) { s = eidx[e]; d = eidx[EE + e]; }
  else        { s = e - EE; d = s; }
  float alpha[HH];
  #pragma unroll
  for (int hh = 0; hh < HH; ++hh)
    alpha[hh] = aBuf[(size_t)e * HH + hh] / denom[d * HH + hh] * 0.25f; // head mean folded
  const float* xls = xl + (size_t)s * HC;
  float a0 = 0.0f, a1 = 0.0f;
  #pragma unroll
  for (int hh = 0; hh < HH; ++hh) {
    a0 += alpha[hh] * xls[hh * CC + lane];
    a1 += alpha[hh] * xls[hh * CC + lane + 32];
  }
  atomicAdd(&accOut[(size_t)d * CC + lane],      a0);
  atomicAdd(&accOut[(size_t)d * CC + lane + 32], a1);
}

// -------------------------------------------------- epilogue: bias (+ residual)
__global__ void epilogue_kernel(const float* __restrict__ acc,
                                const float* __restrict__ bias,
                                const float* __restrict__ resid,
                                float* __restrict__ out) {
  int t = threadIdx.x;
  int c = t & 63;
  int n = blockIdx.x * 4 + (t >> 6);
  float v = acc[(size_t)n * CC + c] + bias[c];
  if (resid) v += resid[(size_t)n * CC + c];
  out[(size_t)n * CC + c] = v;
}

// ============================================================== host launcher
extern "C" void kernel_launch(void* const* d_in, const int* in_sizes, int n_in,
                              void* d_out, int out_size, void* d_ws, size_t ws_size,
                              hipStream_t stream) {
  (void)in_sizes; (void)n_in; (void)out_size; (void)ws_size;
  const float*     x     = (const float*)d_in[0];
  const long long* eidx  = (const long long*)d_in[1];
  const float*     ea    = (const float*)d_in[2];
  const long long* batch = (const long long*)d_in[3];
  const float*     Wl    = (const float*)d_in[4];
  const float*     Wr    = (const float*)d_in[5];
  const float*     We    = (const float*)d_in[6];
  const float*     att   = (const float*)d_in[7];
  const float*     gbias = (const float*)d_in[8];
  const float*     gnw   = (const float*)d_in[9];
  const float*     gnb   = (const float*)d_in[10];
  const float*     gnms  = (const float*)d_in[11];

  char* ws = (char*)d_ws;
  float* hbuf  = (float*)(ws + 0);                     // N*64
  float* xlbuf = (float*)(ws + 12800000);              // N*256
  float* xrbuf = (float*)(ws + 64000000);              // N*256
  float* accb  = (float*)(ws + 115200000);             // N*64
  float* xcur  = (float*)(ws + 128000000);             // N*64
  float* abuf  = (float*)(ws + 140800000);             // E2*4 (logits then a)
  float* mbuf  = (float*)(ws + 148000000);             // N*4
  float* dbuf  = (float*)(ws + 148800000);             // N*4
  float* S1    = (float*)(ws + 149600000);             // B*64
  float* S2    = (float*)(ws + 149616384);             // B*64
  float* cnt   = (float*)(ws + 149632768);             // B
  float* Am    = (float*)(ws + 149633024);             // B*64
  float* Bc    = (float*)(ws + 149649408);             // B*64
  float* easum = (float*)(ws + 149665792);             // 16

  // edge_attr mean (sums; consumers divide by E) — constant across blocks
  hipMemsetAsync(easum, 0, DEE * sizeof(float), stream);
  ea_sum_kernel<<<128, 256, 0, stream>>>(ea, easum);

  const float* xin = x;
  for (int L = 0; L < 2; ++L) {
    const float* WlL  = Wl   + (size_t)L * CC * HC;
    const float* WrL  = Wr   + (size_t)L * CC * HC;
    const float* WeL  = We   + (size_t)L * DEE * HC;
    const float* attL = att  + (size_t)L * HC;
    const float* gbL  = gbias+ (size_t)L * CC;
    const float* gnwL = gnw  + (size_t)L * CC;
    const float* gnbL = gnb  + (size_t)L * CC;
    const float* gnmsL= gnms + (size_t)L * CC;

    // zero/prepare scratch for this block
    hipMemsetAsync(S1, 0, (2 * BB * CC + BB) * sizeof(float), stream); // S1,S2,cnt
    hipMemsetAsync(dbuf, 0, NN * HH * sizeof(float), stream);
    hipMemsetAsync(accb, 0, (size_t)NN * CC * sizeof(float), stream);
    fill_neg_inf_kernel<<<(NN * HH + 255) / 256, 256, 0, stream>>>(mbuf, NN * HH);

    // GraphNorm
    stats_kernel<<<NN / 4, 256, 0, stream>>>(xin, batch, S1, S2, cnt);
    finalize_stats_kernel<<<(BB * CC + 255) / 256, 256, 0, stream>>>(S1, S2, cnt,
                                                                     gnwL, gnmsL, Am, Bc);
    normalize_kernel<<<NN / 4, 256, 0, stream>>>(xin, batch, Am, Bc, gnbL, hbuf);

    // Node transforms via fp32 WMMA: xl = h@Wl, xr = h@Wr
    gemm_node_kernel<<<dim3(NN / 16, 4, 2), dim3(32, 4), 0, stream>>>(hbuf, WlL, WrL,
                                                                      xlbuf, xrbuf);

    // GATv2 edge phase
    edge_logits_kernel<<<EE2 / 8, dim3(32, 8), 0, stream>>>(eidx, ea, easum, xlbuf, xrbuf,
                                                            WeL, attL, abuf, mbuf);
    edge_exp_kernel<<<(EE2 * HH + 255) / 256, 256, 0, stream>>>(eidx, mbuf, abuf, dbuf);
    edge_aggregate_kernel<<<EE2 / 8, dim3(32, 8), 0, stream>>>(eidx, abuf, dbuf, xlbuf, accb);

    // bias (+ residual on last block)
    if (L == 1)
      epilogue_kernel<<<NN / 4, 256, 0, stream>>>(accb, gbL, x, (float*)d_out);
    else
      epilogue_kernel<<<NN / 4, 256, 0, stream>>>(accb, gbL, nullptr, xcur);
    xin = xcur;
  }
}